// YOLOLoss_13116830122568
// MI455X (gfx1250) — compile-verified
//
#include <hip/hip_runtime.h>

// ---------------- problem constants (from reference) ----------------
constexpr int   kNB         = 2;                 // boxes per cell
constexpr int   kNC         = 20;                // classes
constexpr int   kPredStride = kNB * 5 + kNC;     // 30 floats per cell
constexpr int   kTgtStride  = 5 + kNC;           // 25 floats per cell
constexpr float kLambdaCoord = 5.0f;
constexpr float kLambdaNoobj = 0.5f;

constexpr int kBlock  = 256;                     // 8 wave32 per workgroup
constexpr int kWaves  = kBlock / 32;
constexpr int kBlocks = 1024;                    // d_ws usage: kBlocks*4 floats = 16 KB

typedef __attribute__((ext_vector_type(2))) float v2f;
typedef __attribute__((ext_vector_type(8))) float v8f;

// ---------------------------------------------------------------------------
// Exact 32-lane fp32 sum using two V_WMMA_F32_16X16X4_F32.
//   W1: A = ones(16x4), B(v0=p, v1=0)  -> D1[m][n] = p_n + p_{n+16} = s_n (all rows equal)
//   W2: A(v0=D1[0], v1=0), B = ones    -> D2[m][n] = 2*s_m   (indexed along M)
//   Sum the 8 D2 VGPRs (= 2*sum of this half's rows), exchange with lane^16, halve.
// EXEC must be all ones: callers invoke this only from convergent code.
// ---------------------------------------------------------------------------
__device__ __forceinline__ float wave_sum_wmma(float p) {
  v2f ones; ones[0] = 1.0f; ones[1] = 1.0f;
  v2f b;    b[0]    = p;    b[1]    = 0.0f;
  v8f c = {};
  v8f d1 = __builtin_amdgcn_wmma_f32_16x16x4_f32(false, ones, false, b,
                                                 (short)0, c, false, false);
  v2f a2; a2[0] = d1[0]; a2[1] = 0.0f;
  v8f d2 = __builtin_amdgcn_wmma_f32_16x16x4_f32(false, a2, false, ones,
                                                 (short)0, c, false, false);
  float h = ((d2[0] + d2[1]) + (d2[2] + d2[3])) +
            ((d2[4] + d2[5]) + (d2[6] + d2[7]));          // = 2 * (sum of 8 rows)
  float o = __shfl_xor(h, 16, 32);                        // other half's rows
  return 0.5f * (h + o);
}

__device__ __forceinline__ float box_iou(float px, float py, float pw, float ph,
                                         float tx, float ty, float tw, float th) {
  float p1x = px - 0.5f * pw, p1y = py - 0.5f * ph;
  float p2x = px + 0.5f * pw, p2y = py + 0.5f * ph;
  float t1x = tx - 0.5f * tw, t1y = ty - 0.5f * th;
  float t2x = tx + 0.5f * tw, t2y = ty + 0.5f * th;
  float ix = fmaxf(fminf(p2x, t2x) - fmaxf(p1x, t1x), 0.0f);
  float iy = fmaxf(fminf(p2y, t2y) - fmaxf(p1y, t1y), 0.0f);
  float inter = ix * iy;
  float pa = (p2x - p1x) * (p2y - p1y);
  float ta = (t2x - t1x) * (t2y - t1y);
  return inter / (pa + ta - inter + 1e-6f);
}

__device__ __forceinline__ float safe_sqrt(float x) {
  return x > 0.0f ? sqrtf(x) : 0.0f;
}

// ------------------------- pass 1: per-block partials -----------------------
__global__ __launch_bounds__(kBlock) void yolo_partial_kernel(
    const float* __restrict__ pred, const float* __restrict__ tgt,
    float* __restrict__ partials, int ncells) {
  float acc_coord = 0.f, acc_cobj = 0.f, acc_cnoobj = 0.f, acc_cls = 0.f;
  const int stride = gridDim.x * blockDim.x;
  for (int cell = blockIdx.x * blockDim.x + threadIdx.x; cell < ncells;
       cell += stride) {
    // Prefetch next grid-stride window (speculative; OOB silently dropped).
    __builtin_prefetch(pred + (size_t)(cell + stride) * kPredStride, 0, 1);
    __builtin_prefetch(tgt  + (size_t)(cell + stride) * kTgtStride,  0, 1);

    // Per-lane contiguous chunks: a wave covers contiguous 3840 B (pred) and 3200 B (tgt) regions,
    // every fetched cacheline is fully consumed.
    const float2* p2 = (const float2*)(pred + (size_t)cell * kPredStride); // 8B aligned (120B stride)
    const float*  t  = tgt + (size_t)cell * kTgtStride;

    float fp[kPredStride];
#pragma unroll
    for (int i = 0; i < kPredStride / 2; ++i) {
      float2 v = p2[i]; fp[2 * i] = v.x; fp[2 * i + 1] = v.y;
    }
    float ft[kTgtStride];
#pragma unroll
    for (int i = 0; i < kTgtStride; ++i) ft[i] = t[i];

    const float tx = ft[0], ty = ft[1], tw = ft[2], th = ft[3], tconf = ft[4];
    const float obj   = (tconf > 0.0f)  ? 1.0f : 0.0f;
    const float noobj = (tconf == 0.0f) ? 1.0f : 0.0f;

    const float iou0 = box_iou(fp[0], fp[1], fp[2], fp[3], tx, ty, tw, th);
    const float iou1 = box_iou(fp[5], fp[6], fp[7], fp[8], tx, ty, tw, th);
    const bool pick1 = iou1 > iou0;        // argmax: first max wins ties
    const float sx = pick1 ? fp[5] : fp[0];
    const float sy = pick1 ? fp[6] : fp[1];
    const float sw = pick1 ? fp[7] : fp[2];
    const float sh = pick1 ? fp[8] : fp[3];
    const float sc = pick1 ? fp[9] : fp[4];

    const float dx = sx - tx, dy = sy - ty;
    const float dw = safe_sqrt(sw) - sqrtf(tw);
    const float dh = safe_sqrt(sh) - sqrtf(th);
    acc_coord  += obj * ((dx * dx + dy * dy) + (dw * dw + dh * dh));
    const float dc = sc - tconf;
    acc_cobj   += obj * dc * dc;
    acc_cnoobj += noobj * (fp[4] * fp[4] + fp[9] * fp[9]);
    float cls = 0.f;
#pragma unroll
    for (int k = 0; k < kNC; ++k) {
      float d = fp[10 + k] - ft[5 + k];
      cls += d * d;
    }
    acc_cls += obj * cls;
  }

  // Convergent from here on: EXEC all ones for WMMA.
  const float r0 = wave_sum_wmma(acc_coord);
  const float r1 = wave_sum_wmma(acc_cobj);
  const float r2 = wave_sum_wmma(acc_cnoobj);
  const float r3 = wave_sum_wmma(acc_cls);

  __shared__ float sdata[kWaves][4];
  const unsigned lane = threadIdx.x & 31u;
  const unsigned wave = threadIdx.x >> 5;
  if (lane == 0) {
    sdata[wave][0] = r0; sdata[wave][1] = r1;
    sdata[wave][2] = r2; sdata[wave][3] = r3;
  }
  __syncthreads();
  if (threadIdx.x == 0) {
    float s0 = 0.f, s1 = 0.f, s2 = 0.f, s3 = 0.f;
#pragma unroll
    for (int w = 0; w < kWaves; ++w) {
      s0 += sdata[w][0]; s1 += sdata[w][1];
      s2 += sdata[w][2]; s3 += sdata[w][3];
    }
    float* o = partials + (size_t)blockIdx.x * 4;
    o[0] = s0; o[1] = s1; o[2] = s2; o[3] = s3;
  }
}

// ------------------------- pass 2: deterministic finalize -------------------
__global__ __launch_bounds__(kBlock) void yolo_final_kernel(
    const float* __restrict__ partials, int nparts, float* __restrict__ out) {
  float s0 = 0.f, s1 = 0.f, s2 = 0.f, s3 = 0.f;
  for (int i = threadIdx.x; i < nparts; i += blockDim.x) {
    const float* p = partials + (size_t)i * 4;
    s0 += p[0]; s1 += p[1]; s2 += p[2]; s3 += p[3];
  }
  const float r0 = wave_sum_wmma(s0);
  const float r1 = wave_sum_wmma(s1);
  const float r2 = wave_sum_wmma(s2);
  const float r3 = wave_sum_wmma(s3);

  __shared__ float sdata[kWaves][4];
  const unsigned lane = threadIdx.x & 31u;
  const unsigned wave = threadIdx.x >> 5;
  if (lane == 0) {
    sdata[wave][0] = r0; sdata[wave][1] = r1;
    sdata[wave][2] = r2; sdata[wave][3] = r3;
  }
  __syncthreads();
  if (threadIdx.x == 0) {
    float coord = 0.f, cobj = 0.f, cnoobj = 0.f, cls = 0.f;
#pragma unroll
    for (int w = 0; w < kWaves; ++w) {
      coord += sdata[w][0]; cobj += sdata[w][1];
      cnoobj += sdata[w][2]; cls += sdata[w][3];
    }
    out[0] = kLambdaCoord * coord + cobj + kLambdaNoobj * cnoobj + cls;
    out[1] = coord;
    out[2] = cobj;
    out[3] = cnoobj;
    out[4] = cls;
  }
}

// ---------------------------------------------------------------------------
extern "C" void kernel_launch(void* const* d_in, const int* in_sizes, int n_in,
                              void* d_out, int out_size, void* d_ws, size_t ws_size,
                              hipStream_t stream) {
  const float* pred = (const float*)d_in[0];   // (B,S,S,30) fp32
  const float* tgt  = (const float*)d_in[1];   // (B,S,S,25) fp32
  float* out        = (float*)d_out;           // 5 fp32 scalars
  float* partials   = (float*)d_ws;            // kBlocks*4 floats scratch

  const int ncells = in_sizes[0] / kPredStride;  // B*S*S = 802816

  yolo_partial_kernel<<<kBlocks, kBlock, 0, stream>>>(pred, tgt, partials, ncells);
  yolo_final_kernel<<<1, kBlock, 0, stream>>>(partials, kBlocks, out);
}